// KNNClassifierLayer_71966472011989
// MI455X (gfx1250) — compile-verified
//
#include <hip/hip_runtime.h>
#include <hip/hip_bf16.h>

// KNN classifier: B=1024 queries, N=32768 train points, DIM=64, K_NN=8, C=6.
// Fused fp32-WMMA GEMM + top-8 selection; no materialized [B,N] matrix.

#define B_Q        1024
#define N_TRAIN    32768
#define DIM        64
#define K_NN       8
#define NUM_CLS    6
#define QT         16      // query tile (WMMA M)
#define NT         16      // train tile (WMMA N)
#define NCHUNKS    8
#define CHUNK      (N_TRAIN / NCHUNKS)          // 4096
#define TILES_PER_CHUNK (CHUNK / NT)            // 256
#define WAVES      8

typedef float v2f __attribute__((ext_vector_type(2)));
typedef float v8f __attribute__((ext_vector_type(8)));

__device__ __forceinline__ bool lex_less(float d0, int i0, float d1, int i1) {
    return (d0 < d1) || (d0 == d1 && i0 < i1);
}

// ---------------------------------------------------------------------------
// Kernel 1: per (query-tile, N-chunk) block: WMMA distance tiles + exact
// per-lane top-8, merged to one partial top-8 per query per chunk.
// grid = (64, 8), block = 256 (8 waves, wave32)
// ---------------------------------------------------------------------------
__global__ __launch_bounds__(256)
void knn_partial_kernel(const float* __restrict__ x,
                        const float* __restrict__ xt,
                        float* __restrict__ ws_d,
                        int*   __restrict__ ws_i) {
    __shared__ float lds_tile[WAVES][QT * NT];     // per-wave transpose slab
    __shared__ float md[QT * 16 * K_NN];           // merge: 16 lists x 8 per query
    __shared__ int   mi[QT * 16 * K_NN];

    const int tid  = threadIdx.x;
    const int wave = tid >> 5;
    const int lane = tid & 31;
    const int hi   = lane >> 4;      // half-wave: 0 or 1
    const int lq   = lane & 15;      // row-in-tile (A/M) and col-in-tile (B/N)
    const int m0   = blockIdx.x * QT;
    const int nch0 = blockIdx.y * CHUNK;

    // --- Preload A fragments (x rows, reused for every tile) ---------------
    // WMMA f32 16x16x4 A layout: lanes 0-15 hold M=lane, K={4i,4i+1};
    // lanes 16-31 hold M=lane-16, K={4i+2,4i+3}.
    v2f afrag[16];
    float xsq_part = 0.0f;
    const float* arow = x + (size_t)(m0 + lq) * DIM + 2 * hi;
#pragma unroll
    for (int i = 0; i < 16; ++i) {
        v2f a = *(const v2f*)(arow + 4 * i);
        afrag[i] = a;
        xsq_part += a.x * a.x + a.y * a.y;
    }
    // full ||x_m||^2 for m = lq, replicated across both half-waves
    float xsq_all = xsq_part + __shfl_xor(xsq_part, 16, 32);
    // per-accumulator-VGPR row norms: C/D layout row = v + 8*hi
    float xsq_r[8];
#pragma unroll
    for (int v = 0; v < 8; ++v)
        xsq_r[v] = __shfl(xsq_all, v + 8 * hi, 32);

    // --- lane-local exact top-8 (ascending; [7] = worst) -------------------
    float bd[K_NN]; int bi[K_NN];
#pragma unroll
    for (int j = 0; j < K_NN; ++j) { bd[j] = 3.4e38f; bi[j] = 0x7fffffff; }

    // --- main loop: each wave owns tiles wave, wave+8, ... -----------------
    for (int t = wave; t < TILES_PER_CHUNK; t += WAVES) {
        const int n0 = nch0 + t * NT;
        const float* brow = xt + (size_t)(n0 + lq) * DIM + 2 * hi;
        if (t + WAVES < TILES_PER_CHUNK)
            __builtin_prefetch(brow + (size_t)WAVES * NT * DIM, 0, 1);

        v8f acc = {};
        float tsq_part = 0.0f;
#pragma unroll
        for (int i = 0; i < 16; ++i) {
            // B layout mirrors A: lanes 0-15 hold N=lane, K={4i,4i+1};
            // lanes 16-31 hold N=lane-16, K={4i+2,4i+3}.
            v2f b = *(const v2f*)(brow + 4 * i);
            tsq_part += b.x * b.x + b.y * b.y;
            acc = __builtin_amdgcn_wmma_f32_16x16x4_f32(
                false, afrag[i], false, b, (short)0, acc, false, false);
        }
        // full ||t_n||^2 for n = n0 + lq (what every VGPR of this lane needs)
        float tsq = tsq_part + __shfl_xor(tsq_part, 16, 32);

        // d2 = max(||x||^2 + ||t||^2 - 2*cross, 0); transpose via LDS
        float* lt = lds_tile[wave];
#pragma unroll
        for (int v = 0; v < 8; ++v) {
            float d2 = fmaxf(fmaf(-2.0f, acc[v], xsq_r[v] + tsq), 0.0f);
            lt[(v + 8 * hi) * NT + lq] = d2;      // [row m][col n]
        }
        __builtin_amdgcn_wave_barrier();          // wave-private LDS RAW order

        // lane owns query lq, candidate slots n = 8*hi .. 8*hi+7
#pragma unroll
        for (int j = 0; j < 8; ++j) {
            float d  = lt[lq * NT + 8 * hi + j];
            int   id = n0 + 8 * hi + j;
            if (lex_less(d, id, bd[K_NN - 1], bi[K_NN - 1])) {
                bd[K_NN - 1] = d; bi[K_NN - 1] = id;
#pragma unroll
                for (int s = K_NN - 1; s > 0; --s) {
                    if (lex_less(bd[s], bi[s], bd[s - 1], bi[s - 1])) {
                        float td = bd[s]; bd[s] = bd[s - 1]; bd[s - 1] = td;
                        int   ti = bi[s]; bi[s] = bi[s - 1]; bi[s - 1] = ti;
                    }
                }
            }
        }
        __builtin_amdgcn_wave_barrier();
    }

    // --- merge 16 lane-lists per query (8 waves x 2 half-waves) ------------
    const int list_id = wave * 2 + hi;            // 0..15
#pragma unroll
    for (int j = 0; j < K_NN; ++j) {
        md[(lq * 16 + list_id) * K_NN + j] = bd[j];
        mi[(lq * 16 + list_id) * K_NN + j] = bi[j];
    }
    __syncthreads();

    if (tid < QT) {
        const int q = tid;
        // exact top-8 of 128 uniquely-indexed candidates:
        // successive selection with lexicographic (d, idx) ordering.
        float prevd = -1.0f; int previ = -1;
        const int gq = m0 + q;
        const size_t wbase = ((size_t)blockIdx.y * B_Q + gq) * K_NN;
        for (int r = 0; r < K_NN; ++r) {
            float cd = 3.5e38f; int ci = 0x7fffffff;
            for (int u = 0; u < 16 * K_NN; ++u) {
                float d  = md[q * (16 * K_NN) + u];
                int   id = mi[q * (16 * K_NN) + u];
                bool after  = (d > prevd) || (d == prevd && id > previ);
                if (after && lex_less(d, id, cd, ci)) { cd = d; ci = id; }
            }
            prevd = cd; previ = ci;
            ws_d[wbase + r] = cd;
            ws_i[wbase + r] = ci;
        }
    }
}

// ---------------------------------------------------------------------------
// Kernel 2: merge the 8 partial lists per query, histogram labels, normalize.
// one thread per query
// ---------------------------------------------------------------------------
__global__ __launch_bounds__(256)
void knn_finalize_kernel(const float* __restrict__ ws_d,
                         const int*   __restrict__ ws_i,
                         const int*   __restrict__ y_train,
                         float*       __restrict__ out) {
    const int q = blockIdx.x * blockDim.x + threadIdx.x;
    if (q >= B_Q) return;

    int cnt[NUM_CLS];
#pragma unroll
    for (int c = 0; c < NUM_CLS; ++c) cnt[c] = 0;

    float prevd = -1.0f; int previ = -1;
    for (int r = 0; r < K_NN; ++r) {
        float cd = 3.5e38f; int ci = 0x7fffffff;
        for (int c = 0; c < NCHUNKS; ++c) {
            const size_t base = ((size_t)c * B_Q + q) * K_NN;
#pragma unroll
            for (int j = 0; j < K_NN; ++j) {
                float d  = ws_d[base + j];
                int   id = ws_i[base + j];
                bool after = (d > prevd) || (d == prevd && id > previ);
                if (after && lex_less(d, id, cd, ci)) { cd = d; ci = id; }
            }
        }
        prevd = cd; previ = ci;
        const int lab = y_train[ci];
#pragma unroll
        for (int c = 0; c < NUM_CLS; ++c) cnt[c] += (lab == c) ? 1 : 0;
    }
#pragma unroll
    for (int c = 0; c < NUM_CLS; ++c)
        out[(size_t)q * NUM_CLS + c] = (float)cnt[c] * (1.0f / K_NN);
}

// ---------------------------------------------------------------------------
extern "C" void kernel_launch(void* const* d_in, const int* in_sizes, int n_in,
                              void* d_out, int out_size, void* d_ws, size_t ws_size,
                              hipStream_t stream) {
    const float* x  = (const float*)d_in[0];   // [1024, 64]
    const float* xt = (const float*)d_in[1];   // [32768, 64]
    const int*   y  = (const int*)d_in[2];     // [32768]
    float* out = (float*)d_out;                // [1024, 6]

    // workspace: NCHUNKS*B*8 floats + NCHUNKS*B*8 ints = 512 KB
    float* ws_dists = (float*)d_ws;
    int*   ws_idx   = (int*)((char*)d_ws +
                      (size_t)NCHUNKS * B_Q * K_NN * sizeof(float));

    dim3 g1(B_Q / QT, NCHUNKS), b1(256);
    knn_partial_kernel<<<g1, b1, 0, stream>>>(x, xt, ws_dists, ws_idx);

    dim3 g2((B_Q + 255) / 256), b2(256);
    knn_finalize_kernel<<<g2, b2, 0, stream>>>(ws_dists, ws_idx, y, out);
}